// XORNet_87368224735623
// MI455X (gfx1250) — compile-verified
//
#include <hip/hip_runtime.h>

// CDNA5 / gfx1250 SNN (snntorch Leaky, reset-by-subtraction) forward pass.
//
// Roofline: 8MB read + 80MB written = ~88MB @ 23.3 TB/s -> ~3.8us floor.
// VALU: ~490 ops/row (20 unrolled LIF steps) -> comparable; so we minimize
// inner-loop ops (cmp reused for spike+next reset, cndmask between
// precomputed cur / cur-1) and keep 4 independent chains for dual-issue.
// The front-end GEMM cur = x @ w1^T (K=2,N=4) runs on V_WMMA_F32_16X16X4_F32,
// with a one-shot per-wave LDS transpose from the WMMA C-layout into
// per-lane row ownership.

typedef __attribute__((ext_vector_type(2))) float v2f;
typedef __attribute__((ext_vector_type(8))) float v8f;

#define SNN_BETA 0.9f

__global__ __launch_bounds__(256) void snn_lif_kernel(
    const float* __restrict__ x,   // [B,2]
    const float* __restrict__ w1,  // [4,2]
    const float* __restrict__ w2,  // [1,4]
    float* __restrict__ out,       // [20,B]
    int B) {
  __shared__ __align__(16) float lds_cur[256 * 4];  // per-block cur tile

  const int tid      = threadIdx.x;
  const int lane     = tid & 31;
  const int waveBase = tid & ~31;               // block-local base row of wave
  const int row      = blockIdx.x * 256 + tid;  // global batch row owned

  // ---------------- cur = x @ w1^T via V_WMMA_F32_16X16X4_F32 ----------------
  // A (16x4): VGPR0 = {lanes0-15: K0, lanes16-31: K2}, VGPR1 = {K1, K3}.
  // We have K=2, so lanes 16-31 carry zeros. Two WMMAs cover 32 rows/wave.
  const int gbase = blockIdx.x * 256 + waveBase;
  const int l15   = lane & 15;
  v2f g0 = *(const v2f*)(x + 2 * (gbase + l15));        // rows waveBase+0..15
  v2f g1 = *(const v2f*)(x + 2 * (gbase + 16 + l15));   // rows waveBase+16..31

  const bool lo = lane < 16;
  v2f aA, aB;
  aA.x = lo ? g0.x : 0.0f;  aA.y = lo ? g0.y : 0.0f;
  aB.x = lo ? g1.x : 0.0f;  aB.y = lo ? g1.y : 0.0f;

  // B (4x16): col n (= lane, n<4) holds w1 row n; K0 in VGPR0, K1 in VGPR1.
  {
  }
  const int  hsel = lane & 3;
  const float bw0 = w1[2 * hsel + 0];
  const float bw1 = w1[2 * hsel + 1];
  const bool act  = lane < 4;
  v2f bmat;
  bmat.x = act ? bw0 : 0.0f;
  bmat.y = act ? bw1 : 0.0f;

  v8f czero = {};
  // 8 args: (neg_a, A, neg_b, B, c_mod, C, reuse_a, reuse_b)
  v8f d0 = __builtin_amdgcn_wmma_f32_16x16x4_f32(
      false, aA, false, bmat, (short)0, czero, false, false);
  v8f d1 = __builtin_amdgcn_wmma_f32_16x16x4_f32(
      false, aB, false, bmat, (short)0, czero, false, false);

  // C/D layout: VGPR r -> (lanes0-15: M=r, N=lane) (lanes16-31: M=8+r, N=lane-16)
  // Scatter cols<4 into [row][h] in LDS, then each lane gathers its own row.
  const int col   = lane & 15;
  const int rhalf = (lane >> 4) * 8;
  if (col < 4) {
#pragma unroll
    for (int r = 0; r < 8; ++r) {
      lds_cur[(waveBase + rhalf + r) * 4 + col]      = d0[r];
      lds_cur[(waveBase + 16 + rhalf + r) * 4 + col] = d1[r];
    }
  }
  // Same-wave DS ops are in-order; region is wave-private, no barrier needed.
  float4 cur4 = *(const float4*)&lds_cur[tid * 4];

  // ---------------- 20 unrolled LIF steps, all state in registers -----------
  const float c0 = cur4.x, c1 = cur4.y, c2 = cur4.z, c3 = cur4.w;
  const float d0c = c0 - 1.0f, d1c = c1 - 1.0f, d2c = c2 - 1.0f, d3c = c3 - 1.0f;
  const float W0 = w2[0], W1 = w2[1], W2 = w2[2], W3 = w2[3];

  float m10 = 0.0f, m11 = 0.0f, m12 = 0.0f, m13 = 0.0f, m2 = 0.0f;
  bool  p10 = false, p11 = false, p12 = false, p13 = false, p2 = false;

  float* op = out + row;
#pragma unroll
  for (int t = 0; t < 20; ++t) {
    // m1 = beta*m1 + cur - prev_spike ; prev predicate doubles as reset
    m10 = SNN_BETA * m10 + (p10 ? d0c : c0);
    m11 = SNN_BETA * m11 + (p11 ? d1c : c1);
    m12 = SNN_BETA * m12 + (p12 ? d2c : c2);
    m13 = SNN_BETA * m13 + (p13 ? d3c : c3);
    p10 = m10 > 1.0f;  // spk1 this step == reset1 next step
    p11 = m11 > 1.0f;
    p12 = m12 > 1.0f;
    p13 = m13 > 1.0f;
    // out = spk1 . w2  (spikes are 0/1 -> masked add, no muls)
    float o = (p10 ? W0 : 0.0f) + (p11 ? W1 : 0.0f) +
              (p12 ? W2 : 0.0f) + (p13 ? W3 : 0.0f);
    m2 = SNN_BETA * m2 + (p2 ? (o - 1.0f) : o);
    p2 = m2 > 1.0f;
    op[(size_t)t * B] = p2 ? 1.0f : 0.0f;  // coalesced b32 store per step
  }
}

extern "C" void kernel_launch(void* const* d_in, const int* in_sizes, int n_in,
                              void* d_out, int out_size, void* d_ws, size_t ws_size,
                              hipStream_t stream) {
  const float* x  = (const float*)d_in[0];  // [B,2]
  const float* w1 = (const float*)d_in[1];  // [4,2]
  const float* w2 = (const float*)d_in[2];  // [1,4]
  float* out = (float*)d_out;               // [20,B,1]

  const int B = in_sizes[0] / 2;            // 1,048,576
  const int blocks = B / 256;               // B is a multiple of 256 (exact cover
                                            // keeps EXEC all-1s for WMMA)
  snn_lif_kernel<<<blocks, 256, 0, stream>>>(x, w1, w2, out, B);
}